// wvfn_23845658427502
// MI455X (gfx1250) — compile-verified
//
#include <hip/hip_runtime.h>

typedef __attribute__((ext_vector_type(2))) float v2f;
typedef __attribute__((ext_vector_type(8))) float v8f;

__device__ __forceinline__ float frcp(float x) { return __builtin_amdgcn_rcpf(x); }
__device__ __forceinline__ float frsq(float x) { return __builtin_amdgcn_rsqf(x); }
__device__ __forceinline__ float fexp(float x) { return __builtin_amdgcn_exp2f(x * 1.4426950408889634f); }
// cross-lane gather (wave32): dst[lane] = x[srcLane]
__device__ __forceinline__ float lane_gather(float x, int srcLane) {
    return __int_as_float(__builtin_amdgcn_ds_bpermute(srcLane << 2, __float_as_int(x)));
}

// One walker per lane. Six complex-weighted reductions over n=0..3 are done as
// D[6,16] = W[6,4] x E[4,16] via V_WMMA_F32_16X16X4_F32 (two WMMAs per wave32:
// walker groups [0..15] and [16..31]). Everything is straight-line so EXEC
// stays all-ones through the WMMAs; only the final stores are predicated.
__global__ __launch_bounds__(256) void hydrogenic_hpsi_kernel(
    const float* __restrict__ Rs, const float* __restrict__ A,
    const float* __restrict__ C, float* __restrict__ out, int W)
{
    const int tid  = blockIdx.x * blockDim.x + threadIdx.x;
    const int w    = (tid < W) ? tid : (W - 1);     // clamp tail, keep EXEC full
    const int lane = threadIdx.x & 31;
    const bool lo  = lane < 16;

    // ---- load 12 coords (48 B, 16B-aligned) as 3x float4 -------------------
    const float4* p = reinterpret_cast<const float4*>(Rs + (size_t)w * 12);
    float4 q0 = p[0], q1 = p[1], q2 = p[2];
    float x0=q0.x, y0=q0.y, z0=q0.z;
    float x1=q0.w, y1=q1.x, z1=q1.y;
    float x2=q1.z, y2=q1.w, z2=q2.x;
    float x3=q2.y, y3=q2.z, z3=q2.w;

    // ---- radii: r_k and 1/r_k from a single v_rsq each ---------------------
    float r20 = x0*x0 + y0*y0 + z0*z0;
    float r21 = x1*x1 + y1*y1 + z1*z1;
    float r22 = x2*x2 + y2*y2 + z2*z2;
    float r23 = x3*x3 + y3*y3 + z3*z3;
    float ir0 = frsq(r20), ir1 = frsq(r21), ir2 = frsq(r22), ir3 = frsq(r23);
    float rsum    = r20*ir0 + r21*ir1 + r22*ir2 + r23*ir3;
    float invrsum = ir0 + ir1 + ir2 + ir3;

    // ---- e_n = exp(-rsum / A_n) --------------------------------------------
    float e0 = fexp(-rsum * frcp(A[0]));
    float e1 = fexp(-rsum * frcp(A[1]));
    float e2 = fexp(-rsum * frcp(A[2]));
    float e3 = fexp(-rsum * frcp(A[3]));

    // ---- all-pairs Coulomb: V = -VB * sum 1/|r_a - r_b|  (VB = 1) ----------
    float V = 0.0f;
    {
        float dx, dy, dz, d2;
        dx=x0-x1; dy=y0-y1; dz=z0-z1; d2=dx*dx+dy*dy+dz*dz; V += frsq(d2);
        dx=x0-x2; dy=y0-y2; dz=z0-z2; d2=dx*dx+dy*dy+dz*dz; V += frsq(d2);
        dx=x0-x3; dy=y0-y3; dz=z0-z3; d2=dx*dx+dy*dy+dz*dz; V += frsq(d2);
        dx=x1-x2; dy=y1-y2; dz=z1-z2; d2=dx*dx+dy*dy+dz*dz; V += frsq(d2);
        dx=x1-x3; dy=y1-y3; dz=z1-z3; d2=dx*dx+dy*dy+dz*dz; V += frsq(d2);
        dx=x2-x3; dy=y2-y3; dz=z2-z3; d2=dx*dx+dy*dy+dz*dz; V += frsq(d2);
        V = -V;
    }

    // ---- WMMA A-operand: constant weight matrix W[16,4] (rows m=0..5 used) -
    // row m: 0:Re(C_n) 1:Im(C_n) 2:Re(C_n)a_n 3:Im(C_n)a_n 4:Re(C_n)b_n 5:Im(C_n)b_n
    // with a_n = 4/A_n^2, b_n = 2/A_n.  A-layout (16x4 f32): lane L -> M=L&15,
    // VGPR0 holds K = (L<16 ? 0 : 2), VGPR1 holds K+1.
    const int m  = lane & 15;
    const int k0 = (lane >> 4) << 1;                 // 0 or 2
    float aL = A[k0],     aH = A[k0 + 1];
    float crL = C[2*k0],     ciL = C[2*k0 + 1];
    float crH = C[2*k0 + 2], ciH = C[2*k0 + 3];
    float alL = 4.0f * frcp(aL*aL), beL = 2.0f * frcp(aL);
    float alH = 4.0f * frcp(aH*aH), beH = 2.0f * frcp(aH);
    v2f Aop;
    {
        float v = 0.0f;
        v = (m==0)? crL     : v;  v = (m==1)? ciL     : v;
        v = (m==2)? crL*alL : v;  v = (m==3)? ciL*alL : v;
        v = (m==4)? crL*beL : v;  v = (m==5)? ciL*beL : v;
        Aop.x = v;
        v = 0.0f;
        v = (m==0)? crH     : v;  v = (m==1)? ciH     : v;
        v = (m==2)? crH*alH : v;  v = (m==3)? ciH*alH : v;
        v = (m==4)? crH*beH : v;  v = (m==5)? ciH*beH : v;
        Aop.y = v;
    }

    // ---- WMMA B-operands: E[4,16] per walker group -------------------------
    // B-layout (4x16 f32): lane L -> N=L&15, VGPR0 holds K=(L<16?0:2), VGPR1 K+1.
    const int n0 = lane & 15;        // source lane for group 0 (walkers 0..15)
    const int n1 = n0 + 16;          // source lane for group 1 (walkers 16..31)
    float g0e0 = lane_gather(e0, n0), g0e1 = lane_gather(e1, n0);
    float g0e2 = lane_gather(e2, n0), g0e3 = lane_gather(e3, n0);
    float g1e0 = lane_gather(e0, n1), g1e1 = lane_gather(e1, n1);
    float g1e2 = lane_gather(e2, n1), g1e3 = lane_gather(e3, n1);
    v2f B0, B1;
    B0.x = lo ? g0e0 : g0e2;  B0.y = lo ? g0e1 : g0e3;
    B1.x = lo ? g1e0 : g1e2;  B1.y = lo ? g1e1 : g1e3;

    v8f Cz = {0.f,0.f,0.f,0.f,0.f,0.f,0.f,0.f};
    // 8 args: (neg_a, A, neg_b, B, c_mod, C, reuse_a, reuse_b)
    v8f D0 = __builtin_amdgcn_wmma_f32_16x16x4_f32(false, Aop, false, B0,
                                                   (short)0, Cz, false, false);
    v8f D1 = __builtin_amdgcn_wmma_f32_16x16x4_f32(false, Aop, false, B1,
                                                   (short)0, Cz, false, false);

    // ---- redistribute D: row m, col N sits in VGPR m at lane N (m<8) -------
    float s[6];
#pragma unroll
    for (int i = 0; i < 6; ++i) {
        float d1 = lane_gather(D1[i], lane & 15);
        s[i] = lo ? D0[i] : d1;
    }
    float psiRe = s[0], psiIm = s[1];
    float aRe   = s[2], aIm   = s[3];
    float bRe   = s[4], bIm   = s[5];

    // ---- H psi, observables -------------------------------------------------
    float nabRe = aRe - invrsum * bRe;
    float nabIm = aIm - invrsum * bIm;
    float HRe = -0.5f * nabRe + V * psiRe;
    float HIm = -0.5f * nabIm + V * psiIm;
    float outRe = psiRe * HRe + psiIm * HIm;   // conj(psi)*Hpsi, VB^2 = 1
    float outIm = psiRe * HIm - psiIm * HRe;
    float prob  = psiRe * psiRe + psiIm * psiIm;

    if (tid < W) {
        out[2*(size_t)tid]     = outRe;
        out[2*(size_t)tid + 1] = outIm;
        out[2*(size_t)W + tid] = prob;
    }
}

extern "C" void kernel_launch(void* const* d_in, const int* in_sizes, int n_in,
                              void* d_out, int out_size, void* d_ws, size_t ws_size,
                              hipStream_t stream) {
    const float* Rs = (const float*)d_in[0];   // [W,4,3] f32
    const float* A  = (const float*)d_in[1];   // [4] f32
    const float* C  = (const float*)d_in[2];   // [4] complex64 -> 8 f32
    float* out = (float*)d_out;                // [2W complex floats][W prob floats]
    const int W = in_sizes[0] / 12;
    const int threads = 256;
    const int blocks  = (W + threads - 1) / threads;
    hydrogenic_hpsi_kernel<<<blocks, threads, 0, stream>>>(Rs, A, C, out, W);
}